// WordEmbedding_83227876262331
// MI455X (gfx1250) — compile-verified
//
#include <hip/hip_runtime.h>
#include <hip/hip_bf16.h>

// WordEmbedding: out[b,s,:] = weight[tensor[b,s], :]
// Pure gather, memory-bound. DIM = 512 floats = 2 KB per row.
// One wave32 per row: 4 x 16B vectors per lane, fully coalesced.

#define EMB_DIM     512
#define VEC_PER_ROW (EMB_DIM / 4)       // 128 x 16B vectors per row

// Native clang vector type: valid for __builtin_nontemporal_store and
// lowers to global_load_b128 / global_store_b128.
typedef float v4f __attribute__((ext_vector_type(4)));

__global__ __launch_bounds__(256) void WordEmbedding_gather_kernel(
    const int* __restrict__ idx,      // [n_rows] token ids
    const float* __restrict__ weight, // [VOCAB, 512]
    float* __restrict__ out,          // [n_rows, 512]
    int n_rows) {
  const int gtid = blockIdx.x * blockDim.x + threadIdx.x;
  const int row  = gtid >> 5;         // one wave (32 lanes) per row
  const int lane = gtid & 31;
  if (row >= n_rows) return;

  const int tok = idx[row];           // wave-uniform -> scalarized load

  const v4f* __restrict__ src =
      reinterpret_cast<const v4f*>(weight) + (size_t)tok * VEC_PER_ROW;
  v4f* __restrict__ dst =
      reinterpret_cast<v4f*>(out) + (size_t)row * VEC_PER_ROW;

  // Issue all 4 b128 loads first (4 outstanding LOADcnt), then NT stores.
  v4f r0 = src[lane + 0 * 32];
  v4f r1 = src[lane + 1 * 32];
  v4f r2 = src[lane + 2 * 32];
  v4f r3 = src[lane + 3 * 32];

  // Output is write-once: non-temporal keeps the 64MB result from evicting
  // weight rows out of the 192MB L2.
  __builtin_nontemporal_store(r0, &dst[lane + 0 * 32]);
  __builtin_nontemporal_store(r1, &dst[lane + 1 * 32]);
  __builtin_nontemporal_store(r2, &dst[lane + 2 * 32]);
  __builtin_nontemporal_store(r3, &dst[lane + 3 * 32]);
}

extern "C" void kernel_launch(void* const* d_in, const int* in_sizes, int n_in,
                              void* d_out, int out_size, void* d_ws, size_t ws_size,
                              hipStream_t stream) {
  const int*   idx    = (const int*)d_in[0];    // tensor [16, 2048]
  const float* weight = (const float*)d_in[1];  // [32000, 512]
  float*       out    = (float*)d_out;          // [16, 2048, 512]

  const int n_rows = in_sizes[0];               // 16 * 2048 = 32768 tokens

  // 256 threads = 8 wave32 = 8 rows per block
  const int rows_per_block = 256 / 32;
  const int grid = (n_rows + rows_per_block - 1) / rows_per_block;

  WordEmbedding_gather_kernel<<<grid, 256, 0, stream>>>(idx, weight, out, n_rows);
}